// CoarseMatching_609885356631
// MI455X (gfx1250) — compile-verified
//
#include <hip/hip_runtime.h>
#include <hip/hip_bf16.h>

typedef float v2f __attribute__((ext_vector_type(2)));
typedef float v8f __attribute__((ext_vector_type(8)));

#define NB 2
#define LL 4800
#define CC 256
#define GRID_W 80
#define BIG ((size_t)LL * (size_t)LL)

// ---------------------------------------------------------------------------
// K0: zero the reduction scratch (rowsum/colsum float, rowmax/colmax uint bits)
// ---------------------------------------------------------------------------
__global__ void k_init(float* __restrict__ rowsum, float* __restrict__ colsum,
                       unsigned* __restrict__ rowmax, unsigned* __restrict__ colmax) {
  const int i = blockIdx.x * 256 + threadIdx.x;
  if (i < NB * LL) {
    rowsum[i] = 0.0f;
    colsum[i] = 0.0f;
    rowmax[i] = 0u;
    colmax[i] = 0u;
  }
}

// ---------------------------------------------------------------------------
// K1: sim = (f0 . f1^T) / (C * TEMP), fp32 WMMA 16x16x4.
// One wave -> 64x64 output tile (4x4 register block of 16x16 WMMA tiles).
// A layout (16x4 f32): lanes 0-15 hold K={0,1}, lanes 16-31 hold K={2,3},
// VGPR0=K even, VGPR1=K odd -> one float2 load per matrix per K-step.
// ---------------------------------------------------------------------------
__global__ __launch_bounds__(256) void k_gemm(const float* __restrict__ f0,
                                              const float* __restrict__ f1,
                                              float* __restrict__ sim) {
  const int wave = (blockIdx.x * 256 + threadIdx.x) >> 5;
  const int lane = threadIdx.x & 31;
  const int TPB = LL / 64;        // 75 tiles per side
  const int WPB = TPB * TPB;      // 5625 tiles per batch
  if (wave >= NB * WPB) return;   // uniform per wave -> EXEC stays all-1s
  const int n = wave / WPB;
  const int t = wave - n * WPB;
  const int l0 = (t / TPB) * 64;
  const int s0 = (t % TPB) * 64;
  const int r = lane & 15;
  const int h = lane >> 4;

  const float* __restrict__ A = f0 + (size_t)n * LL * CC;
  const float* __restrict__ B = f1 + (size_t)n * LL * CC;
  const float* pa[4];
  const float* pb[4];
#pragma unroll
  for (int i = 0; i < 4; ++i) {
    pa[i] = A + (size_t)(l0 + 16 * i + r) * CC + 2 * h;
    pb[i] = B + (size_t)(s0 + 16 * i + r) * CC + 2 * h;
  }

  const v8f vzero = {0.0f, 0.0f, 0.0f, 0.0f, 0.0f, 0.0f, 0.0f, 0.0f};
  v8f acc[4][4];
#pragma unroll
  for (int i = 0; i < 4; ++i)
#pragma unroll
    for (int j = 0; j < 4; ++j) acc[i][j] = vzero;

  for (int c = 0; c < CC; c += 4) {
    v2f a[4], b[4];
#pragma unroll
    for (int i = 0; i < 4; ++i) a[i] = *(const v2f*)(pa[i] + c);
#pragma unroll
    for (int j = 0; j < 4; ++j) b[j] = *(const v2f*)(pb[j] + c);
#pragma unroll
    for (int i = 0; i < 4; ++i)
#pragma unroll
      for (int j = 0; j < 4; ++j)
        acc[i][j] = __builtin_amdgcn_wmma_f32_16x16x4_f32(
            false, a[i], false, b[j], (short)0, acc[i][j], false, false);
  }

  // sim = dot / (sqrt(C)^2 * TEMP) = dot / 25.6
  const float scale = 1.0f / 25.6f;
  float* out = sim + (size_t)n * BIG;
  // D layout: VGPR v -> row M=v (lanes 0-15) or M=v+8 (lanes 16-31), col = lane&15
#pragma unroll
  for (int i = 0; i < 4; ++i)
#pragma unroll
    for (int j = 0; j < 4; ++j) {
      const int col = s0 + 16 * j + r;
#pragma unroll
      for (int v = 0; v < 8; ++v) {
        const int row = l0 + 16 * i + 8 * h + v;
        out[(size_t)row * LL + col] = acc[i][j][v] * scale;
      }
    }
}

// ---------------------------------------------------------------------------
// K2: rowsum[l] = sum_s exp(sim), colsum[s] = sum_l exp(sim).
// Thread owns a column (coalesced reads); colsum in register, rowsum via
// 5-step wave32 shuffle reduce + one atomicAdd per wave per row.
// ---------------------------------------------------------------------------
__global__ __launch_bounds__(256) void k_sums(const float* __restrict__ sim,
                                              float* __restrict__ rowsum,
                                              float* __restrict__ colsum) {
  const int n = blockIdx.z;
  const int s = blockIdx.x * 256 + threadIdx.x;
  const bool sv = s < LL;
  const int lane = threadIdx.x & 31;
  const int l0 = blockIdx.y * (LL / 16);  // 300-row chunk
  const float* __restrict__ base = sim + (size_t)n * BIG;

  float csum = 0.0f;
  for (int l = l0; l < l0 + LL / 16; ++l) {
    float e = sv ? __expf(base[(size_t)l * LL + s]) : 0.0f;
    csum += e;
    float rs = e;
#pragma unroll
    for (int m = 16; m >= 1; m >>= 1) rs += __shfl_xor(rs, m, 32);
    if (lane == 0) atomicAdd(&rowsum[n * LL + l], rs);
  }
  if (sv) atomicAdd(&colsum[n * LL + s], csum);
}

// ---------------------------------------------------------------------------
// K3: conf = exp(sim)^2 / (rowsum[l] * colsum[s]), written in place over sim.
// Tracks rowmax/colmax of conf as uint bit patterns (conf >= 0).
// ---------------------------------------------------------------------------
__global__ __launch_bounds__(256) void k_conf(float* __restrict__ simconf,
                                              const float* __restrict__ rowsum,
                                              const float* __restrict__ colsum,
                                              unsigned* __restrict__ rowmax,
                                              unsigned* __restrict__ colmax) {
  const int n = blockIdx.z;
  const int s = blockIdx.x * 256 + threadIdx.x;
  const bool sv = s < LL;
  const int lane = threadIdx.x & 31;
  const int l0 = blockIdx.y * (LL / 16);
  float* base = simconf + (size_t)n * BIG;

  const float inv_cs = sv ? 1.0f / colsum[n * LL + s] : 0.0f;
  float cmax = 0.0f;
  for (int l = l0; l < l0 + LL / 16; ++l) {
    const float inv_rs = 1.0f / rowsum[n * LL + l];  // uniform -> scalar load
    float c = 0.0f;
    if (sv) {
      const float e = __expf(base[(size_t)l * LL + s]);
      c = (e * e) * (inv_rs * inv_cs);
      base[(size_t)l * LL + s] = c;
      cmax = fmaxf(cmax, c);
    }
    float rm = c;
#pragma unroll
    for (int m = 16; m >= 1; m >>= 1) rm = fmaxf(rm, __shfl_xor(rm, m, 32));
    if (lane == 0) atomicMax(&rowmax[n * LL + l], __float_as_uint(rm));
  }
  if (sv) atomicMax(&colmax[n * LL + s], __float_as_uint(cmax));
}

// ---------------------------------------------------------------------------
// K4: mask = (conf > 0.2) & border-valid(l) & border-valid(s)
//            & (conf == rowmax) & (conf == colmax)     [bit-exact compare]
// ---------------------------------------------------------------------------
__global__ __launch_bounds__(256) void k_mask(const float* __restrict__ conf,
                                              const unsigned* __restrict__ rowmax,
                                              const unsigned* __restrict__ colmax,
                                              float* __restrict__ mask) {
  const int n = blockIdx.z;
  const int l = blockIdx.y;
  const int s = blockIdx.x * 256 + threadIdx.x;
  if (s >= LL) return;
  const size_t idx = (size_t)n * BIG + (size_t)l * LL + s;
  const float c = conf[idx];
  const unsigned cb = __float_as_uint(c);
  const int ly = l / GRID_W, lx = l - ly * GRID_W;
  const int sy = s / GRID_W, sx = s - sy * GRID_W;
  const bool v = (c > 0.2f) &&
                 (ly >= 2) && (ly < 58) && (lx >= 2) && (lx < 78) &&
                 (sy >= 2) && (sy < 58) && (sx >= 2) && (sx < 78) &&
                 (cb == rowmax[n * LL + l]) && (cb == colmax[n * LL + s]);
  mask[idx] = v ? 1.0f : 0.0f;
}

// ---------------------------------------------------------------------------
extern "C" void kernel_launch(void* const* d_in, const int* in_sizes, int n_in,
                              void* d_out, int out_size, void* d_ws, size_t ws_size,
                              hipStream_t stream) {
  (void)in_sizes; (void)n_in; (void)out_size; (void)ws_size;
  const float* f0 = (const float*)d_in[0];  // [2,4800,256] f32
  const float* f1 = (const float*)d_in[1];  // [2,4800,256] f32

  float* conf = (float*)d_out;              // conf region; also stages sim
  float* mask = conf + (size_t)NB * BIG;    // mask region (floats 0/1)

  float* rowsum = (float*)d_ws;             // [2*4800]
  float* colsum = rowsum + NB * LL;         // [2*4800]
  unsigned* rowmax = (unsigned*)(colsum + NB * LL);  // [2*4800]
  unsigned* colmax = rowmax + NB * LL;               // [2*4800]

  k_init<<<(NB * LL + 255) / 256, 256, 0, stream>>>(rowsum, colsum, rowmax, colmax);

  const int waves = NB * (LL / 64) * (LL / 64);       // 11250 tile-waves
  k_gemm<<<(waves + 7) / 8, 256, 0, stream>>>(f0, f1, conf);

  dim3 g2((LL + 255) / 256, 16, NB);                  // 19 col-chunks x 16 row-chunks
  k_sums<<<g2, 256, 0, stream>>>(conf, rowsum, colsum);
  k_conf<<<g2, 256, 0, stream>>>(conf, rowsum, colsum, rowmax, colmax);

  dim3 g4((LL + 255) / 256, LL, NB);
  k_mask<<<g4, 256, 0, stream>>>(conf, rowmax, colmax, mask);
}